// AttentiveAggregation_78563541778578
// MI455X (gfx1250) — compile-verified
//
#include <hip/hip_runtime.h>
#include <cstdint>

#define NV    1000000   // nodes
#define NB    32768     // segments
#define ND    256       // feature dim
#define CHUNK 16        // rows staged per iteration (= WMMA M)
#define ROW_F 268       // padded LDS row stride in floats: bank-conflict-free A-gather, 16B aligned rows
#define ROW_B (ROW_F * 4)

typedef __attribute__((ext_vector_type(2))) float v2f;
typedef __attribute__((ext_vector_type(4))) float v4f;
typedef __attribute__((ext_vector_type(8))) float v8f;

// Build segment boundaries from sorted batch ids: seg_start[b] = first row with batch >= b,
// seg_start[NB] = NV. Every entry is written exactly once.
__global__ __launch_bounds__(256) void seg_bounds_kernel(
    const int* __restrict__ batch, int* __restrict__ seg_start)
{
    int v = blockIdx.x * blockDim.x + threadIdx.x;
    if (v > NV) return;
    int bc = (v == NV) ? NB : batch[v];
    int bp = (v == 0) ? -1 : batch[v - 1];
    for (int b = bp + 1; b <= bc; ++b) seg_start[b] = v;
}

// One wave (32 lanes) per segment. Fused: async-stage H rows to LDS, WMMA scores,
// online softmax, weighted pooling. H is read from HBM exactly once.
// Note: Wb shifts all scores by a constant -> cancels in softmax -> dropped (exact).
__global__ __launch_bounds__(64) void attn_pool_kernel(
    const float* __restrict__ H, const float* __restrict__ Ww,
    const int* __restrict__ seg_start, float* __restrict__ out)
{
    __shared__ __align__(16) float s_w[ND];
    __shared__ __align__(16) float s_buf[2][CHUNK * ROW_F];

    const int tid = threadIdx.x;
    // stage weight vector (64 threads x 4 floats = 256)
    *(v4f*)&s_w[tid * 4] = *(const v4f*)&Ww[tid * 4];
    __syncthreads();

    const int wave = tid >> 5;
    const int lane = tid & 31;
    const int seg  = blockIdx.x * 2 + wave;

    const int r0 = seg_start[seg];
    const int n  = seg_start[seg + 1] - r0;

    float* buf = s_buf[wave];
    const uint32_t buf_lds = (uint32_t)(uintptr_t)buf;  // LDS byte offset (low 32 bits of AS3 ptr)
    const int row_a = lane & 15;   // WMMA A: M index held by this lane
    const int hi    = lane >> 4;   // WMMA A: lanes 16-31 hold K=2,3 of each 4-wide K tile

    float m = -INFINITY, sum = 0.f;
    v4f acc0 = {0.f, 0.f, 0.f, 0.f};
    v4f acc1 = {0.f, 0.f, 0.f, 0.f};

    for (int base = 0; base < n; base += CHUNK) {
        const int cnt = min(CHUNK, n - base);

        // ---- async stage cnt rows (1KB each) into this wave's LDS buffer ----
        for (int i = 0; i < cnt; ++i) {
            const float* src = H + (size_t)(r0 + base + i) * ND + lane * 4;  // 16B per lane
            uint32_t dst = buf_lds + i * ROW_B + lane * 16;
            asm volatile("global_load_async_to_lds_b128 %0, %1, off th:TH_LOAD_NT"
                         :: "v"(dst), "v"(src) : "memory");
            const float* src2 = src + 128;   // second 512B half of the row
            uint32_t dst2 = dst + 512;
            asm volatile("global_load_async_to_lds_b128 %0, %1, off th:TH_LOAD_NT"
                         :: "v"(dst2), "v"(src2) : "memory");
        }
        asm volatile("s_wait_asynccnt 0" ::: "memory");

        // ---- scores via V_WMMA_F32_16X16X4_F32: D[m][*] = sum_k Hrow_m[k] * w[k] ----
        // A 16x4 layout: lane L holds row L%16, VGPR pair = K {2*(L/16), 2*(L/16)+1}.
        // B 4x16: weight pair broadcast to all 16 columns (mirrored K layout).
        v8f c = {};
        #pragma unroll
        for (int t = 0; t < ND / 4; ++t) {
            const int kk = 4 * t + 2 * hi;
            v2f a = *(const v2f*)&buf[row_a * ROW_F + kk];
            v2f b = *(const v2f*)&s_w[kk];
            c = __builtin_amdgcn_wmma_f32_16x16x4_f32(false, a, false, b,
                                                      (short)0, c, false, false);
        }
        float cs[8];
        #pragma unroll
        for (int j = 0; j < 8; ++j) cs[j] = c[j];   // cs[j] = score of row (j + 8*hi)

        // ---- chunk max over valid rows (scores identical across D columns) ----
        float mloc = -INFINITY;
        #pragma unroll
        for (int j = 0; j < 8; ++j) {
            int r = j + 8 * hi;
            mloc = (r < cnt) ? fmaxf(mloc, cs[j]) : mloc;
        }
        float cmax = fmaxf(mloc, __shfl_xor(mloc, 16, 32));

        // ---- online softmax rescale ----
        float mnew  = fmaxf(m, cmax);
        float scale = __expf(m - mnew);   // first chunk: exp(-inf) == 0
        sum *= scale;
        acc0 *= scale;
        acc1 *= scale;
        m = mnew;

        // ---- weighted accumulation: lane owns features [lane*8, lane*8+8) ----
        #pragma unroll
        for (int r = 0; r < CHUNK; ++r) {
            if (r < cnt) {
                float sr = __shfl(cs[r & 7], (r & 8) ? 16 : 0, 32);
                float e  = __expf(sr - m);
                sum += e;
                const v4f* hp = (const v4f*)&buf[r * ROW_F + lane * 8];
                v4f h0 = hp[0], h1 = hp[1];
                acc0 += e * h0;
                acc1 += e * h1;
            }
        }
    }

    const float rs = (n > 0) ? (1.0f / sum) : 0.0f;   // empty segments -> zeros
    v4f o0 = acc0 * rs;
    v4f o1 = acc1 * rs;

    float* op = out + (size_t)seg * ND + lane * 8;
    __builtin_nontemporal_store(o0, (v4f*)op);
    __builtin_nontemporal_store(o1, (v4f*)(op + 4));
}

extern "C" void kernel_launch(void* const* d_in, const int* in_sizes, int n_in,
                              void* d_out, int out_size, void* d_ws, size_t ws_size,
                              hipStream_t stream)
{
    const float* H     = (const float*)d_in[0];
    const int*   batch = (const int*)d_in[1];
    const float* Ww    = (const float*)d_in[2];
    // d_in[3] (Wb): constant shift of scores -> softmax-invariant -> unused (exact).
    float* out = (float*)d_out;
    int* seg_start = (int*)d_ws;   // (NB+1) ints of scratch

    seg_bounds_kernel<<<((NV + 1) + 255) / 256, 256, 0, stream>>>(batch, seg_start);
    attn_pool_kernel<<<NB / 2, 64, 0, stream>>>(H, Ww, seg_start, out);
}